// FeatureSimilarity_4355096838959
// MI455X (gfx1250) — compile-verified
//
#include <hip/hip_runtime.h>
#include <hip/hip_bf16.h>

typedef __attribute__((ext_vector_type(2))) float v2f;
typedef __attribute__((ext_vector_type(8))) float v8f;
typedef __attribute__((ext_vector_type(4))) int v4i;

#define KDIM 512
#define TILE 128          // output tile per block (128x128)
#define KC   32           // K chunk staged in LDS
#define LDP  36           // padded LDS row stride in floats (144B, 16B aligned)
#define NCHUNK (KDIM / KC)

// Probe for the gfx1250 async global->LDS builtins.
#if defined(__has_builtin)
#if __has_builtin(__builtin_amdgcn_global_load_async_to_lds_b128)
#define HAVE_ASYNC_LDS 1
#endif
#endif
#ifndef HAVE_ASYNC_LDS
#define HAVE_ASYNC_LDS 0
#endif

#if HAVE_ASYNC_LDS
// Parameter types per hipcc diagnostic: arg0 is addrspace(1) int4* ("__device__"),
// arg1 the LDS-side int4* (addrspace 3).
typedef __attribute__((address_space(1))) v4i glb_v4i;
typedef __attribute__((address_space(3))) v4i lds_v4i;
#endif

// ---------------------------------------------------------------------------
// Kernel 1: per-row squared norms. One wave (32 lanes) per row.
// ---------------------------------------------------------------------------
__global__ __launch_bounds__(256) void FeatureSim_norms_kernel(
    const float* __restrict__ F, float* __restrict__ nrm, int nrows) {
  int row  = blockIdx.x * 8 + (threadIdx.x >> 5);
  int lane = threadIdx.x & 31;
  if (row >= nrows) return;
  const float* r = F + (size_t)row * KDIM;
  float s = 0.0f;
#pragma unroll
  for (int k = 0; k < KDIM / 32; ++k) {
    float x = r[lane + k * 32];
    s = fmaf(x, x, s);
  }
#pragma unroll
  for (int off = 16; off > 0; off >>= 1) s += __shfl_xor(s, off, 32);
  if (lane == 0) nrm[row] = s;
}

// ---------------------------------------------------------------------------
// Stage one 128x32 f32 tile pair (A rows, B rows) into LDS.
// Each of the 256 threads moves 16 floats of A and 16 floats of B (4x b128).
// ---------------------------------------------------------------------------
__device__ __forceinline__ void stage_chunk(const float* __restrict__ gA,
                                            const float* __restrict__ gB,
                                            float* sA, float* sB,
                                            int k0, int lr, int lc) {
#if HAVE_ASYNC_LDS
#pragma unroll
  for (int i = 0; i < 4; ++i) {
    __builtin_amdgcn_global_load_async_to_lds_b128(
        (glb_v4i*)(gA + k0 + lc + i * 4),
        (lds_v4i*)(sA + lr * LDP + lc + i * 4), 0, 0);
    __builtin_amdgcn_global_load_async_to_lds_b128(
        (glb_v4i*)(gB + k0 + lc + i * 4),
        (lds_v4i*)(sB + lr * LDP + lc + i * 4), 0, 0);
  }
#else
#pragma unroll
  for (int i = 0; i < 4; ++i) {
    *(float4*)&sA[lr * LDP + lc + i * 4] = *(const float4*)&gA[k0 + lc + i * 4];
    *(float4*)&sB[lr * LDP + lc + i * 4] = *(const float4*)&gB[k0 + lc + i * 4];
  }
#endif
}

__device__ __forceinline__ void wait_stage_done() {
#if HAVE_ASYNC_LDS
  asm volatile("s_wait_asynccnt 0x0" ::: "memory");
#endif
  __syncthreads();
}

// ---------------------------------------------------------------------------
// Kernel 2: tiled f32 WMMA GEMM (G = F F^T) fused with the distance epilogue
//   out[i][j] = -sqrt(max(n[i] + n[j] - 2*G[i][j], 0))
// Block: 256 threads = 8 waves arranged 4 (M) x 2 (N).
// Wave subtile: 32 (M) x 64 (N) = 2x4 WMMA accumulators of 16x16.
// Double-buffered LDS with async global->LDS prefetch of the next K chunk.
// ---------------------------------------------------------------------------
__global__ __launch_bounds__(256) void FeatureSim_pdist_kernel(
    const float* __restrict__ F, const float* __restrict__ nrm,
    float* __restrict__ out, int N) {
  __shared__ float lds[2][2][TILE * LDP];   // [buffer][0=A rows, 1=B rows]

  const int t    = threadIdx.x;
  const int wave = t >> 5;
  const int lane = t & 31;
  const int half = lane >> 4;        // 0: lanes 0-15, 1: lanes 16-31
  const int l15  = lane & 15;

  const int wm = wave & 3;           // wave M position (0..3) -> 32 rows each
  const int wn = wave >> 2;          // wave N position (0..1) -> 64 cols each

  const int rowBase = blockIdx.y * TILE;   // global row tile start
  const int colBase = blockIdx.x * TILE;   // global col tile start

  // Staging assignment: 2 threads per LDS row, 16 floats (4x b128) each.
  const int lr = t >> 1;             // LDS row 0..127
  const int lc = (t & 1) * 16;       // float offset 0 or 16 within K-chunk

  v8f acc[2][4];
#pragma unroll
  for (int mt = 0; mt < 2; ++mt)
#pragma unroll
    for (int nt = 0; nt < 4; ++nt)
      acc[mt][nt] = {};

  const float* gA = F + (size_t)(rowBase + lr) * KDIM;
  const float* gB = F + (size_t)(colBase + lr) * KDIM;

  // Prologue: prefetch chunk 0 into buffer 0.
  stage_chunk(gA, gB, &lds[0][0][0], &lds[0][1][0], 0, lr, lc);

  for (int c = 0; c < NCHUNK; ++c) {
    // Chunk c is resident once our async loads drain and all waves arrive.
    // The same barrier proves every wave finished reading buffer (c+1)&1
    // during iteration c-1, so it is safe to overwrite next.
    wait_stage_done();

    if (c + 1 < NCHUNK)
      stage_chunk(gA, gB, &lds[(c + 1) & 1][0][0], &lds[(c + 1) & 1][1][0],
                  (c + 1) * KC, lr, lc);

    const float* sA = &lds[c & 1][0][0];
    const float* sB = &lds[c & 1][1][0];

#pragma unroll
    for (int ks = 0; ks < KC / 4; ++ks) {
      const int kk = ks * 4 + 2 * half;   // K pair handled by this half-wave
      v2f a[2], b[4];
#pragma unroll
      for (int mt = 0; mt < 2; ++mt)
        a[mt] = *(const v2f*)&sA[(wm * 32 + mt * 16 + l15) * LDP + kk];
#pragma unroll
      for (int nt = 0; nt < 4; ++nt)
        b[nt] = *(const v2f*)&sB[(wn * 64 + nt * 16 + l15) * LDP + kk];
#pragma unroll
      for (int mt = 0; mt < 2; ++mt)
#pragma unroll
        for (int nt = 0; nt < 4; ++nt)
          acc[mt][nt] = __builtin_amdgcn_wmma_f32_16x16x4_f32(
              /*neg_a=*/false, a[mt], /*neg_b=*/false, b[nt],
              /*c_mod=*/(short)0, acc[mt][nt],
              /*reuse_a=*/false, /*reuse_b=*/false);
    }
  }

  // Epilogue: distance transform + store.
  // C/D layout: VGPR v, lanes 0-15 -> M=v, N=lane; lanes 16-31 -> M=8+v, N=lane-16.
#pragma unroll
  for (int mt = 0; mt < 2; ++mt) {
    float ni[8];
#pragma unroll
    for (int v = 0; v < 8; ++v)
      ni[v] = nrm[rowBase + wm * 32 + mt * 16 + half * 8 + v];
#pragma unroll
    for (int nt = 0; nt < 4; ++nt) {
      const int col = colBase + wn * 64 + nt * 16 + l15;
      const float nj = nrm[col];
      const v8f c = acc[mt][nt];
#pragma unroll
      for (int v = 0; v < 8; ++v) {
        const int rowi = rowBase + wm * 32 + mt * 16 + half * 8 + v;
        float d = fmaxf(ni[v] + nj - 2.0f * c[v], 0.0f);
        out[(size_t)rowi * (size_t)N + (size_t)col] = -__builtin_sqrtf(d);
      }
    }
  }
}

extern "C" void kernel_launch(void* const* d_in, const int* in_sizes, int n_in,
                              void* d_out, int out_size, void* d_ws, size_t ws_size,
                              hipStream_t stream) {
  const float* F = (const float*)d_in[0];
  float* out = (float*)d_out;
  float* nrm = (float*)d_ws;             // 8192 floats of scratch

  const int N = in_sizes[0] / KDIM;      // 8192

  // Row squared norms: 8 rows per 256-thread block.
  FeatureSim_norms_kernel<<<dim3((N + 7) / 8), dim3(256), 0, stream>>>(F, nrm, N);

  // Fused GEMM + distance epilogue: 128x128 tile per block.
  dim3 grid(N / TILE, N / TILE);
  FeatureSim_pdist_kernel<<<grid, dim3(256), 0, stream>>>(F, nrm, out, N);
}